// SMCSampler_71159018160559
// MI455X (gfx1250) — compile-verified
//
#include <hip/hip_runtime.h>

#define NP 262144
#define PITCH 68

typedef __attribute__((ext_vector_type(2))) float v2f;
typedef __attribute__((ext_vector_type(8))) float v8f;

__device__ __forceinline__ v8f wmma4(v2f a, v2f b, v8f c) {
  // D = A(16x4 f32) x B(4x16 f32) + C(16x16 f32)
  return __builtin_amdgcn_wmma_f32_16x16x4_f32(false, a, false, b, (short)0, c,
                                               false, false);
}

// ---------------- resampling pipeline ----------------

__global__ __launch_bounds__(256) void k_blockmax(const float* __restrict__ lw,
                                                  float* __restrict__ bmax) {
  __shared__ float s[256];
  int t = threadIdx.x;
  int base = blockIdx.x * 512;
  float m = fmaxf(lw[base + t], lw[base + 256 + t]);
  s[t] = m;
  __syncthreads();
  for (int off = 128; off > 0; off >>= 1) {
    if (t < off) s[t] = fmaxf(s[t], s[t + off]);
    __syncthreads();
  }
  if (t == 0) bmax[blockIdx.x] = s[0];
}

__global__ __launch_bounds__(256) void k_finalmax(const float* __restrict__ bmax,
                                                  float* __restrict__ scal) {
  __shared__ float s[256];
  int t = threadIdx.x;
  s[t] = fmaxf(bmax[t], bmax[t + 256]);
  __syncthreads();
  for (int off = 128; off > 0; off >>= 1) {
    if (t < off) s[t] = fmaxf(s[t], s[t + off]);
    __syncthreads();
  }
  if (t == 0) scal[0] = s[0];
}

__global__ __launch_bounds__(256) void k_expscan(const float* __restrict__ lw,
                                                 const float* __restrict__ scal,
                                                 float* __restrict__ cdf,
                                                 float* __restrict__ bsum) {
  __shared__ float s[256];
  int t = threadIdx.x;
  int i = blockIdx.x * 256 + t;
  float e = expf(lw[i] - scal[0]);
  s[t] = e;
  __syncthreads();
  for (int off = 1; off < 256; off <<= 1) {
    float v = (t >= off) ? s[t - off] : 0.f;
    __syncthreads();
    s[t] += v;
    __syncthreads();
  }
  cdf[i] = s[t];
  if (t == 255) bsum[blockIdx.x] = s[255];
}

__global__ __launch_bounds__(256) void k_scanbsum(const float* __restrict__ bsum,
                                                  float* __restrict__ bpre,
                                                  float* __restrict__ scal) {
  __shared__ float s[256];
  int t = threadIdx.x;
  float l[4];
  float run = 0.f;
  for (int j = 0; j < 4; ++j) {
    l[j] = run;
    run += bsum[4 * t + j];
  }
  s[t] = run;
  __syncthreads();
  for (int off = 1; off < 256; off <<= 1) {
    float v = (t >= off) ? s[t - off] : 0.f;
    __syncthreads();
    s[t] += v;
    __syncthreads();
  }
  float excl = s[t] - run;
  for (int j = 0; j < 4; ++j) bpre[4 * t + j] = excl + l[j];
  if (t == 255) scal[1] = s[255];
}

__global__ __launch_bounds__(256) void k_addoff(float* __restrict__ cdf,
                                                const float* __restrict__ bpre) {
  int i = blockIdx.x * 256 + threadIdx.x;
  cdf[i] += bpre[blockIdx.x];
}

__global__ __launch_bounds__(256) void k_search(const float* __restrict__ cdf,
                                                const float* __restrict__ scal,
                                                const float* __restrict__ u,
                                                int* __restrict__ anc) {
  int i = blockIdx.x * 256 + threadIdx.x;
  float S = scal[1];
  float target = ((float)i + u[0]) * (S * (1.0f / (float)NP));
  int lo = 0, hi = NP;
  while (lo < hi) {
    int mid = (lo + hi) >> 1;
    if (cdf[mid] < target)
      lo = mid + 1;
    else
      hi = mid;
  }
  anc[i] = (lo < NP - 1) ? lo : NP - 1;
}

// ---------------- main fused particle kernel ----------------
// 128 threads = 4 waves; each wave owns 16 rows; block = 64 particles.

__global__ __launch_bounds__(128) void k_main(
    const float* __restrict__ particles, const float* __restrict__ obs,
    const float* __restrict__ eps, const float* __restrict__ Amat,
    const float* __restrict__ Cmat, const int* __restrict__ anc,
    float* __restrict__ out) {
  __shared__ float sA[64 * PITCH];   // A row-major (B = A^T frags contiguous)
  __shared__ float sC[64 * PITCH];   // C row-major
  __shared__ float Sg[64 * PITCH];   // staging: x_res -> m_trans -> innov -> eps -> x_prop
  __shared__ float sObs[64];

  const int t = threadIdx.x;
  const int lane = t & 31;
  const int wave = t >> 5;
  const int m = lane & 15;   // row within slab (A) / col within tile (B,C,D)
  const int hi = lane >> 4;  // lane-half selector
  const int row0 = blockIdx.x * 64;
  float* myS = Sg + wave * 16 * PITCH;

  // ---- P0: cooperative loads ----
#pragma unroll
  for (int r = 0; r < 8; ++r) {
    int q = t + r * 128;  // float4 index 0..1023
    int rr = q >> 4, c4 = (q & 15) << 2;
    *(float4*)&sA[rr * PITCH + c4] = ((const float4*)Amat)[q];
    *(float4*)&sC[rr * PITCH + c4] = ((const float4*)Cmat)[q];
  }
  if (t < 64) sObs[t] = obs[t];
  {
    int row = t >> 1, half = t & 1;
    int g = anc[row0 + row];
    const float4* src = (const float4*)(particles + (size_t)g * 64 + half * 32);
    float4* dst = (float4*)&Sg[row * PITCH + half * 32];
#pragma unroll
    for (int r = 0; r < 8; ++r) dst[r] = src[r];
    if (half == 0) out[(size_t)66 * NP + row0 + row] = (float)g;
  }
  __syncthreads();

  float obs_r[4];
#pragma unroll
  for (int nt = 0; nt < 4; ++nt) obs_r[nt] = sObs[nt * 16 + m];

  v2f af[16];
  v8f accT[4], accP[4], accX[4];

  // ---- GEMM1: m_trans = x_res @ A^T ----
#pragma unroll
  for (int kc = 0; kc < 16; ++kc)
    af[kc] = *(const v2f*)&myS[m * PITCH + kc * 4 + hi * 2];
#pragma unroll
  for (int nt = 0; nt < 4; ++nt) {
    v8f acc = {};
#pragma unroll
    for (int kc = 0; kc < 16; ++kc) {
      v2f bf = *(const v2f*)&sA[(nt * 16 + m) * PITCH + kc * 4 + hi * 2];
      acc = wmma4(af[kc], bf, acc);
    }
    accT[nt] = acc;
  }
  __syncthreads();
#pragma unroll
  for (int nt = 0; nt < 4; ++nt)
#pragma unroll
    for (int v = 0; v < 8; ++v)
      myS[(v + 8 * hi) * PITCH + nt * 16 + m] = accT[nt][v];
  __syncthreads();

  // ---- GEMM2: innov = obs - m_trans @ C^T ----
#pragma unroll
  for (int kc = 0; kc < 16; ++kc)
    af[kc] = *(const v2f*)&myS[m * PITCH + kc * 4 + hi * 2];
#pragma unroll
  for (int nt = 0; nt < 4; ++nt) {
    v8f acc = {};
#pragma unroll
    for (int kc = 0; kc < 16; ++kc) {
      v2f bf = *(const v2f*)&sC[(nt * 16 + m) * PITCH + kc * 4 + hi * 2];
      acc = wmma4(af[kc], bf, acc);
    }
#pragma unroll
    for (int v = 0; v < 8; ++v) acc[v] = obs_r[nt] - acc[v];
    accP[nt] = acc;  // innov
  }
  __syncthreads();
#pragma unroll
  for (int nt = 0; nt < 4; ++nt)
#pragma unroll
    for (int v = 0; v < 8; ++v)
      myS[(v + 8 * hi) * PITCH + nt * 16 + m] = accP[nt][v];
  __syncthreads();

  // ---- GEMM3: p = innov @ C  (B = C, non-transposed: strided frag loads) ----
#pragma unroll
  for (int kc = 0; kc < 16; ++kc)
    af[kc] = *(const v2f*)&myS[m * PITCH + kc * 4 + hi * 2];
#pragma unroll
  for (int nt = 0; nt < 4; ++nt) {
    v8f acc = {};
#pragma unroll
    for (int kc = 0; kc < 16; ++kc) {
      int k = kc * 4 + hi * 2;
      v2f bf;
      bf.x = sC[k * PITCH + nt * 16 + m];
      bf.y = sC[(k + 1) * PITCH + nt * 16 + m];
      acc = wmma4(af[kc], bf, acc);
    }
    accP[nt] = acc;  // p = innov @ C
  }
  __syncthreads();

  // ---- load eps into staging ----
  {
    int row = t >> 1, half = t & 1;
    const float4* src =
        (const float4*)(eps + (size_t)(row0 + row) * 64 + half * 32);
    float4* dst = (float4*)&Sg[row * PITCH + half * 32];
#pragma unroll
    for (int r = 0; r < 8; ++r) dst[r] = src[r];
  }
  __syncthreads();

  // ---- elementwise: x_prop, trans/prop partial sums ----
  float ssT[8] = {0.f}, ssQ[8] = {0.f}, ssE[8] = {0.f};
#pragma unroll
  for (int nt = 0; nt < 4; ++nt) {
    v8f xp;
#pragma unroll
    for (int v = 0; v < 8; ++v) {
      float e = myS[(v + 8 * hi) * PITCH + nt * 16 + m];
      float p = accP[nt][v];
      float d = 0.5f * p + e;          // x_prop - m_trans
      xp[v] = accT[nt][v] + d;         // m_trans + 0.5*p + e
      ssT[v] += d * d;
      ssQ[v] += e * e;
    }
    accX[nt] = xp;
  }
  __syncthreads();
#pragma unroll
  for (int nt = 0; nt < 4; ++nt)
#pragma unroll
    for (int v = 0; v < 8; ++v)
      myS[(v + 8 * hi) * PITCH + nt * 16 + m] = accX[nt][v];
  __syncthreads();

  // ---- GEMM4: y = x_prop @ C^T ; emission residual ----
#pragma unroll
  for (int kc = 0; kc < 16; ++kc)
    af[kc] = *(const v2f*)&myS[m * PITCH + kc * 4 + hi * 2];
#pragma unroll
  for (int nt = 0; nt < 4; ++nt) {
    v8f acc = {};
#pragma unroll
    for (int kc = 0; kc < 16; ++kc) {
      v2f bf = *(const v2f*)&sC[(nt * 16 + m) * PITCH + kc * 4 + hi * 2];
      acc = wmma4(af[kc], bf, acc);
    }
#pragma unroll
    for (int v = 0; v < 8; ++v) {
      float r = obs_r[nt] - acc[v];
      ssE[v] += r * r;
    }
  }

  // ---- per-row reductions: sum over 16 lanes of each half ----
#pragma unroll
  for (int v = 0; v < 8; ++v) {
    float s = ssT[v] + ssE[v] - ssQ[v];
    s += __shfl_xor(s, 1, 32);
    s += __shfl_xor(s, 2, 32);
    s += __shfl_xor(s, 4, 32);
    s += __shfl_xor(s, 8, 32);
    if (m == v) {
      int rg = row0 + wave * 16 + v + 8 * hi;
      // inc = trans_lp + emis_lp - prop_lp = -0.5*s - 32*log(2*pi)
      float inc = -0.5f * s - 58.812066125099054f;
      out[(size_t)65 * NP + rg] = inc;                  // log_w_inc
      out[rg] = inc - 12.476649250079015f;              // -log(N) + inc
    }
  }

  // ---- cooperative store of x_prop ----
  {
    int row = t >> 1, half = t & 1;
    float4* dst =
        (float4*)(out + (size_t)NP + (size_t)(row0 + row) * 64 + half * 32);
    const float4* src = (const float4*)&Sg[row * PITCH + half * 32];
#pragma unroll
    for (int r = 0; r < 8; ++r) dst[r] = src[r];
  }
}

// ---------------- launch ----------------

extern "C" void kernel_launch(void* const* d_in, const int* in_sizes, int n_in,
                              void* d_out, int out_size, void* d_ws,
                              size_t ws_size, hipStream_t stream) {
  const float* log_w = (const float*)d_in[0];
  const float* particles = (const float*)d_in[1];
  const float* observation = (const float*)d_in[2];
  const float* eps = (const float*)d_in[3];
  const float* u = (const float*)d_in[4];
  const float* A = (const float*)d_in[5];
  const float* C = (const float*)d_in[6];
  float* out = (float*)d_out;

  float* wsf = (float*)d_ws;
  float* cdf = wsf;                    // N floats
  int* anc = (int*)(wsf + NP);         // N ints
  float* bmax = wsf + 2 * NP;          // 512
  float* bsum = bmax + 512;            // 1024
  float* bpre = bsum + 1024;           // 1024
  float* scal = bpre + 1024;           // [0]=gmax, [1]=S

  k_blockmax<<<512, 256, 0, stream>>>(log_w, bmax);
  k_finalmax<<<1, 256, 0, stream>>>(bmax, scal);
  k_expscan<<<1024, 256, 0, stream>>>(log_w, scal, cdf, bsum);
  k_scanbsum<<<1, 256, 0, stream>>>(bsum, bpre, scal);
  k_addoff<<<1024, 256, 0, stream>>>(cdf, bpre);
  k_search<<<1024, 256, 0, stream>>>(cdf, scal, u, anc);
  k_main<<<NP / 64, 128, 0, stream>>>(particles, observation, eps, A, C, anc,
                                      out);
}